// SpatialCoRL_39994735461019
// MI455X (gfx1250) — compile-verified
//
#include <hip/hip_runtime.h>

typedef float v2f __attribute__((ext_vector_type(2)));
typedef float v8f __attribute__((ext_vector_type(8)));

#define NB   8
#define CC   128
#define HH   96
#define WW   96
#define PS   9          // patch size (displacements per axis)
#define YT   16         // output rows per tile
#define XT   14         // output cols per tile
#define PR   18         // prod rows needed per tile (YT + 2)
#define PMW  16         // prod cols (WMMA M) per tile (XT + 2)
#define BSTR 10         // LDS stride over dj (padded 9 -> 10)
#define NYT  6          // 96 / 16
#define NXT  7          // X0 in {0,14,...,84}

// out[b,di,dj,yo,xo] = sum_{i,j in 3x3} prod(y'=yo+i, x'=xo+j, dj)
// prod(y',x',dj) = sum_c qpad1[c,y',x'] * kpad5[c, y'+di, x'+dj]
// Band GEMM: D[m,n] = sum_c A[m,c]*B[c,n],
//   A[m,c]=qpad1[c, y', X0+m], B[c,n]=kpad5[c, y'+di, X0+n(+16)]
//   => D[m,n] = prod(x'=X0+m, dj=n-m (+16)), keep 0<=dj<=8.
__global__ __launch_bounds__(256)
void corr_wmma_kernel(const float* __restrict__ Q,
                      const float* __restrict__ K,
                      float* __restrict__ Out)
{
    __shared__ float band[PR * PMW * BSTR];   // [prod_row][m][dj], 11.25 KB

    // decode block id -> (batch, di, ytile, xtile)
    int bid = blockIdx.x;
    const int xt = bid % NXT; bid /= NXT;
    const int yt = bid % NYT; bid /= NYT;
    const int di = bid % PS;  bid /= PS;
    const int bi = bid;

    const int X0 = xt * XT;
    const int Y0 = yt * YT;

    const int tid  = threadIdx.x;
    const int lane = tid & 31;      // wave32
    const int wave = tid >> 5;      // 8 waves
    const int ln   = lane & 15;     // M (A) / N (B) index
    const int g    = lane >> 4;     // lane group: K pair selector

    const size_t chs = (size_t)HH * WW;                 // channel stride (9216)
    const float* Qb = Q + (size_t)bi * CC * chs;
    const float* Kb = K + (size_t)bi * CC * chs;

    // per-lane column validity (loop invariant)
    const int  xq   = X0 + ln - 1;        // q col (unpadded), pad=1
    const bool axok = (xq  >= 0) && (xq  < WW);
    const int  xk0  = X0 + ln - 5;        // k col, pad=5, B tile 0
    const int  xk1  = X0 + 16 + ln - 5;   // B tile 1
    const bool c0ok = (xk0 >= 0) && (xk0 < WW);
    const bool c1ok = (xk1 >= 0) && (xk1 < WW);

    for (int r = wave; r < PR; r += 8) {      // uniform per wave
        const int  yq   = Y0 + r;             // prod row, padded coords 0..97
        const int  qy   = yq - 1;             // q row
        const int  ky   = yq + di - 5;        // k row
        const bool ayok = (qy >= 0) && (qy < HH);
        const bool byok = (ky >= 0) && (ky < HH);

        const bool aok  = ayok && axok;
        const bool bok0 = byok && c0ok;
        const bool bok1 = byok && c1ok;

        // clamped base pointers (in-bounds even for masked lanes), at channel 2*g
        const float* qp  = Qb + (size_t)(2*g) * chs + (size_t)(ayok ? qy : 0) * WW + (axok ? xq  : 0);
        const float* kp0 = Kb + (size_t)(2*g) * chs + (size_t)(byok ? ky : 0) * WW + (c0ok ? xk0 : 0);
        const float* kp1 = Kb + (size_t)(2*g) * chs + (size_t)(byok ? ky : 0) * WW + (c1ok ? xk1 : 0);

        v8f acc0 = {0.f,0.f,0.f,0.f,0.f,0.f,0.f,0.f};
        v8f acc1 = {0.f,0.f,0.f,0.f,0.f,0.f,0.f,0.f};

        #pragma unroll 8
        for (int kk = 0; kk < CC / 4; ++kk) {   // K chunks of 4 channels
            const size_t co = (size_t)kk * 4 * chs;
            // Unconditional loads (addresses are clamped in-bounds), then
            // branchless zero-fill via select -> v_cndmask, no exec toggling.
            const float qa0 = qp [co];
            const float qa1 = qp [co + chs];
            const float kb0 = kp0[co];
            const float kb1 = kp0[co + chs];
            const float kc0 = kp1[co];
            const float kc1 = kp1[co + chs];
            v2f a, b0, b1;
            a.x  = aok  ? qa0 : 0.f;
            a.y  = aok  ? qa1 : 0.f;
            b0.x = bok0 ? kb0 : 0.f;
            b0.y = bok0 ? kb1 : 0.f;
            b1.x = bok1 ? kc0 : 0.f;
            b1.y = bok1 ? kc1 : 0.f;
            acc0 = __builtin_amdgcn_wmma_f32_16x16x4_f32(
                       false, a, false, b0, (short)0, acc0, false, false);
            acc1 = __builtin_amdgcn_wmma_f32_16x16x4_f32(
                       false, a, false, b1, (short)0, acc1, false, false);
        }

        // Band extraction. D layout: VGPR v, lane l -> M = v + 8*(l/16), N = l%16.
        #pragma unroll
        for (int v = 0; v < 8; ++v) {
            const int mm  = v + 8 * g;
            const int dj0 = ln - mm;          // tile0: col = ln
            if (dj0 >= 0 && dj0 <= 8)
                band[(r * PMW + mm) * BSTR + dj0] = acc0[v];
            const int dj1 = ln + 16 - mm;     // tile1: col = ln + 16
            if (dj1 >= 0 && dj1 <= 8)
                band[(r * PMW + mm) * BSTR + dj1] = acc1[v];
        }
    }

    __syncthreads();

    // Phase 2: 3x3 box sums over (y', x') for each dj, write to global.
    const size_t obase = (size_t)(bi * PS + di) * PS * HH * WW;
    for (int idx = tid; idx < PS * YT * XT; idx += 256) {
        const int dj  = idx / (YT * XT);
        const int rem = idx - dj * (YT * XT);
        const int yl  = rem / XT;
        const int xl  = rem - yl * XT;
        float s = 0.f;
        #pragma unroll
        for (int i = 0; i < 3; ++i)
            #pragma unroll
            for (int j = 0; j < 3; ++j)
                s += band[((yl + i) * PMW + (xl + j)) * BSTR + dj];
        const int yo = Y0 + yl;               // always < 96
        const int xo = X0 + xl;               // may reach 97 on last tile
        if (xo < WW)
            Out[obase + ((size_t)dj * HH + yo) * WW + xo] = s;
    }
}

extern "C" void kernel_launch(void* const* d_in, const int* in_sizes, int n_in,
                              void* d_out, int out_size, void* d_ws, size_t ws_size,
                              hipStream_t stream) {
    (void)in_sizes; (void)n_in; (void)out_size; (void)d_ws; (void)ws_size;
    const float* q = (const float*)d_in[0];
    const float* k = (const float*)d_in[1];
    // d_in[2] (value) is unused by the reference output.
    float* out = (float*)d_out;
    const int grid = NB * PS * NYT * NXT;     // 8*9*6*7 = 3024 blocks
    corr_wmma_kernel<<<grid, 256, 0, stream>>>(q, k, out);
}